// DynamicDedispersionLayer_13537736917686
// MI455X (gfx1250) — compile-verified
//
#include <hip/hip_runtime.h>

// Problem constants (match reference)
#define B_N     4
#define D_N     16
#define P_N     2
#define T_N     1024
#define F_N     1024
#define CHUNK_F 128
#define NCHUNK  8          // F_N / CHUNK_F

typedef __attribute__((ext_vector_type(4))) float v4f;

// ---------------------------------------------------------------------------
// Kernel 1: compute delays (second output) and per-(b,d,chunk) integer shifts.
//   delays[b,d,f]   = dm[b,d] * df[f]                     (written to d_out tail)
//   s[b,d,c]        = max(0, trunc(mean(delays[b,d,c*128 : c*128+128])))
// 64 blocks (one per (b,d)), 256 threads each.
// ---------------------------------------------------------------------------
__global__ __launch_bounds__(256) void dedisp_setup_kernel(
    const float* __restrict__ dm_values,   // (4,16)
    const float* __restrict__ disp,        // (1024,)
    float*       __restrict__ out_delays,  // (4,16,1024)
    int*         __restrict__ s_out)       // (4,16,8)
{
    __shared__ float lds[F_N];
    const int bd  = blockIdx.x;            // 0..63  == b*16 + d
    const int tid = threadIdx.x;           // 0..255
    const float dm = dm_values[bd];        // uniform -> scalar load

    #pragma unroll
    for (int k = 0; k < 4; ++k) {
        const int f = tid + k * 256;       // coalesced
        const float v = dm * disp[f];
        out_delays[(size_t)bd * F_N + f] = v;
        lds[f] = v;
    }
    __syncthreads();

    if (tid < NCHUNK) {
        float sum = 0.f;
        #pragma unroll 8
        for (int i = 0; i < CHUNK_F; ++i) sum += lds[tid * CHUNK_F + i];
        const float mean = sum * (1.0f / CHUNK_F);
        int s = (int)truncf(mean);         // trunc toward zero, then clamp >= 0
        s_out[bd * NCHUNK + tid] = (s > 0) ? s : 0;
    }
}

// ---------------------------------------------------------------------------
// Kernel 2: row-rotated block copy.
//   out[b,d,p,t, c*128 + j] = x[b,p, (t + s[b,d,c]) & 1023, c*128 + j]
// grid = B*D*P*NCHUNK*TTILES (= 8192) blocks, 256 threads (8 wave32s).
// Each wave owns one row per iteration: 32 lanes x float4 = 128 floats = 512 B.
// Stores are nontemporal (TH_STORE_NT): 512 MB write-once stream must not
// evict the 32 MB L2-resident source tensor.
// ---------------------------------------------------------------------------
#define TTILES   8
#define ROWS_PER_TILE (T_N / TTILES)   // 128 rows per block

__global__ __launch_bounds__(256) void dedisp_copy_kernel(
    const float* __restrict__ x,       // (4,2,1024,1024)
    const int*   __restrict__ s_in,    // (4,16,8)
    float*       __restrict__ out)     // (4,16,2,1024,1024)
{
    int idx = blockIdx.x;
    const int ttile = idx & (TTILES - 1); idx >>= 3;
    const int chunk = idx & (NCHUNK - 1); idx >>= 3;
    const int p     = idx & (P_N - 1);    idx >>= 1;
    const int d     = idx & (D_N - 1);    idx >>= 4;
    const int b     = idx;                // 0..3

    const int s = s_in[(b * D_N + d) * NCHUNK + chunk];   // uniform -> s_load

    const int lane = threadIdx.x & 31;    // wave32: lane owns 16 B column slice
    const int wave = threadIdx.x >> 5;    // 0..7: wave owns a row

    const float* __restrict__ srcBase =
        x + (size_t)(b * P_N + p) * T_N * F_N + chunk * CHUNK_F + lane * 4;
    float* __restrict__ dstBase =
        out + (size_t)((b * D_N + d) * P_N + p) * T_N * F_N + chunk * CHUNK_F + lane * 4;

    const int t0 = ttile * ROWS_PER_TILE;

    #pragma unroll   // full unroll: 16 b128 loads in flight per wave
    for (int it = 0; it < ROWS_PER_TILE / 8; ++it) {
        const int t    = t0 + it * 8 + wave;
        const int tsrc = (t + s) & (T_N - 1);           // % T, T is pow2
        const v4f v = *reinterpret_cast<const v4f*>(srcBase + (size_t)tsrc * F_N);
        __builtin_nontemporal_store(
            v, reinterpret_cast<v4f*>(dstBase + (size_t)t * F_N));
    }
}

// ---------------------------------------------------------------------------
// Launch. Inputs (setup_inputs order): x, dm_values, dispersion_factor.
// d_out = [dedispersed (4*16*2*1024*1024 f32), delays (4*16*1024 f32)].
// d_ws  = 512 ints of shift values.
// ---------------------------------------------------------------------------
extern "C" void kernel_launch(void* const* d_in, const int* in_sizes, int n_in,
                              void* d_out, int out_size, void* d_ws, size_t ws_size,
                              hipStream_t stream)
{
    const float* x  = (const float*)d_in[0];
    const float* dm = (const float*)d_in[1];
    const float* df = (const float*)d_in[2];

    float* out        = (float*)d_out;
    float* out_delays = out + (size_t)B_N * D_N * P_N * T_N * F_N;
    int*   s_ws       = (int*)d_ws;

    dedisp_setup_kernel<<<B_N * D_N, 256, 0, stream>>>(dm, df, out_delays, s_ws);

    dedisp_copy_kernel<<<B_N * D_N * P_N * NCHUNK * TTILES, 256, 0, stream>>>(
        x, s_ws, out);
}